// LayerNormalNorm_5377299054673
// MI455X (gfx1250) — compile-verified
//
#include <hip/hip_runtime.h>
#include <cstdint>

// ---------------------------------------------------------------------------
// LayerNormalNorm for x[8192 rows][4096], f32.  HBM-bound: 268 MB total
// traffic -> ~11.5us floor @ 23.3 TB/s.  Strategy: read each row from HBM
// exactly once via CDNA5 async global->LDS DMA (double buffered, ASYNCcnt
// pipelined), keep the row in registers across the 3 passes, use hardware
// v_log_f32 / v_exp_f32 for the transcendentals.  Streaming data (x in,
// out) uses non-temporal cache policy so the 192MB L2 keeps weight/bias
// hot instead of churning on a 268MB one-shot stream.
// ---------------------------------------------------------------------------

#define D          4096
#define TPB        256
#define WAVES      (TPB / 32)
#define ROWS_PER_WG 4
#define EPSV       1e-5f
#define EPS_SQRT   0.0031622776601683794f   // sqrt(1e-5)

typedef float f32x4 __attribute__((ext_vector_type(4)));

__device__ __forceinline__ float wave_reduce_add(float v) {
#pragma unroll
  for (int off = 16; off > 0; off >>= 1)
    v += __shfl_xor(v, off, 32);          // wave32 on gfx1250
  return v;
}

// Reduce N per-thread partials across the 256-thread block; result (full sum)
// is returned in v[] to every thread.
template <int N>
__device__ __forceinline__ void block_reduce(float* v, float* red, int tid) {
  const int lane = tid & 31, wv = tid >> 5;
#pragma unroll
  for (int s = 0; s < N; ++s) v[s] = wave_reduce_add(v[s]);
  if (lane == 0) {
#pragma unroll
    for (int s = 0; s < N; ++s) red[wv * N + s] = v[s];
  }
  __syncthreads();
#pragma unroll
  for (int s = 0; s < N; ++s) {
    float acc = 0.f;
#pragma unroll
    for (int w = 0; w < WAVES; ++w) acc += red[w * N + s];
    v[s] = acc;
  }
  __syncthreads();   // red[] is reused by the next reduction
}

__global__ __launch_bounds__(TPB) void lnn_kernel(
    const float* __restrict__ x, const float* __restrict__ weight,
    const float* __restrict__ bias, float* __restrict__ out) {
  // two 16KB row buffers (double buffer) + reduction scratch
  __shared__ __align__(16) float smem[2 * D + WAVES * 8];
  float* red = &smem[2 * D];

  const int tid = threadIdx.x;
  // Low 32 bits of a flat LDS pointer are the wave-relative LDS byte address
  // (ISA 10.2 aperture rules) -> usable as VDST for async-to-LDS ops.
  const uint32_t lds_base = (uint32_t)(uintptr_t)(&smem[0]);

  // Column ownership: col(c, j) = c*1024 + tid*4 + j  (c=0..3, j=0..3)
  // -> every global/LDS access is a coalesced 16B-per-lane b128 op.
  float ws[16], bs[16];
#pragma unroll
  for (int c = 0; c < 4; ++c) {
    f32x4 wv4 = *(const f32x4*)(weight + c * 1024 + tid * 4);   // RT: stays in L2
    f32x4 bv4 = *(const f32x4*)(bias + c * 1024 + tid * 4);
    ws[c * 4 + 0] = wv4.x; ws[c * 4 + 1] = wv4.y;
    ws[c * 4 + 2] = wv4.z; ws[c * 4 + 3] = wv4.w;
    bs[c * 4 + 0] = bv4.x; bs[c * 4 + 1] = bv4.y;
    bs[c * 4 + 2] = bv4.z; bs[c * 4 + 3] = bv4.w;
  }

  const int row0 = blockIdx.x * ROWS_PER_WG;

  // Kick off async DMA of row0 into buffer 0 (4KB per instruction, x4).
  // Non-temporal: x is streamed exactly once, don't retain it in L2.
  {
    const float* src = x + (size_t)row0 * D + tid * 4;
#pragma unroll
    for (int c = 0; c < 4; ++c) {
      uint32_t loff = lds_base + (uint32_t)(c * 4096 + tid * 16);
      asm volatile("global_load_async_to_lds_b128 %0, %1, off th:TH_LOAD_NT"
                   :: "v"(loff), "v"(src + c * 1024) : "memory");
    }
  }

  const float invD = 1.0f / (float)D;

  for (int k = 0; k < ROWS_PER_WG; ++k) {
    const int row = row0 + k;
    const int cur = k & 1;

    if (k + 1 < ROWS_PER_WG) {
      // Prefetch next row into the other buffer, then retire current row's
      // 4 async loads (leave the 4 prefetch loads in flight).
      const float* src = x + (size_t)(row + 1) * D + tid * 4;
      const uint32_t bufoff = lds_base + (uint32_t)((1 - cur) * D * 4);
#pragma unroll
      for (int c = 0; c < 4; ++c) {
        uint32_t loff = bufoff + (uint32_t)(c * 4096 + tid * 16);
        asm volatile("global_load_async_to_lds_b128 %0, %1, off th:TH_LOAD_NT"
                     :: "v"(loff), "v"(src + c * 1024) : "memory");
      }
      asm volatile("s_wait_asynccnt 0x4" ::: "memory");
    } else {
      asm volatile("s_wait_asynccnt 0x0" ::: "memory");
    }
    __syncthreads();   // row data visible to all waves

    const float* buf = &smem[cur * D];

    // ----- pass 1: mean / biased var -------------------------------------
    float xs[16];
    float p1acc[2] = {0.f, 0.f};
#pragma unroll
    for (int c = 0; c < 4; ++c) {
      f32x4 v = *(const f32x4*)(buf + c * 1024 + tid * 4);   // ds_load_b128
      xs[c * 4 + 0] = v.x; xs[c * 4 + 1] = v.y;
      xs[c * 4 + 2] = v.z; xs[c * 4 + 3] = v.w;
      p1acc[0] += (v.x + v.y) + (v.z + v.w);
      p1acc[1] = fmaf(v.x, v.x, fmaf(v.y, v.y, fmaf(v.z, v.z, fmaf(v.w, v.w, p1acc[1]))));
    }
    block_reduce<2>(p1acc, red, tid);
    const float mean = p1acc[0] * invD;
    const float var  = fmaf(-mean, mean, p1acc[1] * invD);
    const float rstd = rsqrtf(var + EPSV);

    // ----- pass 2: Newton-lambda statistics (5 sums, single pass) --------
    // s1=Σ sgn*l1p, Σ xs*d, Σ d, Σ d², Σ xs*d2  (xs*d2 = |xs|*p1)
    float l1p[16];
    float acc[5] = {0.f, 0.f, 0.f, 0.f, 0.f};
#pragma unroll
    for (int i = 0; i < 16; ++i) {
      const float v = (xs[i] - mean) * rstd;
      xs[i] = v;
      const float a   = fabsf(v);
      const float sgn = (v >= 0.f) ? 1.f : -1.f;     // sign with 0 -> +1
      const float oa  = 1.f + a;
      const float lp  = __logf(oa);                  // v_log_f32 (log1p)
      l1p[i] = lp;
      const float d  = fmaf(oa, lp, -a);             // (1+a)l1p - a
      const float p1 = fmaf(oa * lp, lp, -2.f * d);  // (1+a)l1p² - 2d
      acc[0] += sgn * lp;
      acc[1] = fmaf(v, d, acc[1]);
      acc[2] += d;
      acc[3] = fmaf(d, d, acc[3]);
      acc[4] = fmaf(a, p1, acc[4]);
    }
    block_reduce<5>(acc, red, tid);

    const float s1m   = acc[0] * invD;
    const float dvar  = 2.f * acc[1] * invD;
    const float dmean = acc[2] * invD;
    const float mdd   = acc[3] * invD;
    const float mxd2  = acc[4] * invD;
    const float g1    = 0.5f * dvar - s1m;
    // mean((d-d̄)²) = mean(d²) - d̄²
    const float d2var = 2.f * (mxd2 + fmaf(-dmean, dmean, mdd));
    const float g2    = fmaf(-0.5f * dvar, dvar, 0.5f * d2var);
    const float lam   = 1.f - g1 / (g2 + EPS_SQRT);

    // eta depends only on sign(xs): eta(+)=lam, eta(-)=2-lam
    const float eta_p = lam, eta_n = 2.f - lam;
    const float exp_p = eta_p + ((eta_p >= 0.f) ? EPS_SQRT : -EPS_SQRT);
    const float exp_n = eta_n + ((eta_n >= 0.f) ? EPS_SQRT : -EPS_SQRT);
    const float inv_p = 1.f / exp_p, inv_n = 1.f / exp_n;
    const bool  msk_p = fabsf(eta_p) <= EPS_SQRT;
    const bool  msk_n = fabsf(eta_n) <= EPS_SQRT;

    // ----- pass 3: power transform + affine, coalesced NT b128 stores ----
    float* orow = out + (size_t)row * D;
#pragma unroll
    for (int c = 0; c < 4; ++c) {
      float o[4];
#pragma unroll
      for (int j = 0; j < 4; ++j) {
        const int i = c * 4 + j;
        const float v   = xs[i];
        const bool  pos = (v >= 0.f);
        const float sgn = pos ? 1.f : -1.f;
        const float e   = pos ? exp_p : exp_n;
        const float inv = pos ? inv_p : inv_n;
        const bool  m   = pos ? msk_p : msk_n;
        // pow(1+a, e) - 1 == exp(e * log1p(a)) - 1  -> v_exp_f32
        const float t1 = sgn * inv * (__expf(e * l1p[i]) - 1.f);
        const float t2 = sgn * l1p[i];
        o[j] = fmaf(m ? t2 : t1, ws[i], bs[i]);
      }
      f32x4 ov = {o[0], o[1], o[2], o[3]};
      // write-once stream: non-temporal store, don't pollute L2
      __builtin_nontemporal_store(ov, (f32x4*)(orow + c * 1024 + tid * 4));
    }
    __syncthreads();   // all waves done reading buf before it is refilled
  }
}

extern "C" void kernel_launch(void* const* d_in, const int* in_sizes, int n_in,
                              void* d_out, int out_size, void* d_ws, size_t ws_size,
                              hipStream_t stream) {
  const float* x = (const float*)d_in[0];
  const float* w = (const float*)d_in[1];
  const float* b = (const float*)d_in[2];
  float* out = (float*)d_out;

  const int rows = out_size / D;                 // 8192
  const int blocks = rows / ROWS_PER_WG;         // 2048
  lnn_kernel<<<dim3(blocks), dim3(TPB), 0, stream>>>(x, w, b, out);
}